// model__3058016715096
// MI455X (gfx1250) — compile-verified
//
#include <hip/hip_runtime.h>
#include <hip/hip_bf16.h>
#include <math.h>

// Problem constants (from reference)
#define Bn    512
#define Ln    256
#define NNEG  20
#define LNEG  64
#define Dn    100
#define Kn    14
#define NEGTOT (NNEG * LNEG)   // 1280
#define DC    25               // Dn / 4 float4 chunks per row

typedef __attribute__((ext_vector_type(2))) float v2f;
typedef __attribute__((ext_vector_type(8))) float v8f;

struct __align__(16) f4 { float x, y, z, w; };

// ---------------------------------------------------------------------------
// Kernel 1: gathers.  ys[b,d] = mean_l emb[pos_ids[b,l], d]
//           S [b,d] = (1/64) * sum_{n,l} emb[neg_ids[b,n,l], d]
// One block per b.  125 active lanes = 5 groups x 25 float4-chunks: group g
// accumulates every 5th index with global_load_b128 (512B per wave-instr,
// 4x fewer VMEM issues than scalar), then a fixed-order 5-way LDS reduction.
// ---------------------------------------------------------------------------
__global__ __launch_bounds__(128) void k_gather(const int* __restrict__ pos_ids,
                                                const int* __restrict__ neg_ids,
                                                const float* __restrict__ emb,
                                                float* __restrict__ ys,
                                                float* __restrict__ S) {
    __shared__ f4 red[125];
    const int b   = blockIdx.x;
    const int tid = threadIdx.x;
    const int g   = tid / DC;          // 0..4 (group)
    const int c   = tid % DC;          // 0..24 (float4 chunk)
    const f4* emb4 = (const f4*)emb;

    // ---- positive ids -> ys ----
    if (tid < 125) {
        const int* pid = pos_ids + b * Ln;
        f4 acc = {0.f, 0.f, 0.f, 0.f};
        for (int i = g; i < Ln; i += 5) {
            const f4 t = emb4[pid[i] * DC + c];
            acc.x += t.x; acc.y += t.y; acc.z += t.z; acc.w += t.w;
        }
        red[tid] = acc;
    }
    __syncthreads();
    if (tid < DC) {
        f4 s = red[tid];
#pragma unroll
        for (int q = 1; q < 5; ++q) {
            const f4 t = red[tid + DC * q];
            s.x += t.x; s.y += t.y; s.z += t.z; s.w += t.w;
        }
        const float sc = 1.0f / (float)Ln;
        s.x *= sc; s.y *= sc; s.z *= sc; s.w *= sc;
        ((f4*)ys)[b * DC + tid] = s;
    }
    __syncthreads();

    // ---- negative ids -> S ----
    if (tid < 125) {
        const int* nid = neg_ids + b * NEGTOT;
        f4 acc = {0.f, 0.f, 0.f, 0.f};
        for (int i = g; i < NEGTOT; i += 5) {
            const f4 t = emb4[nid[i] * DC + c];
            acc.x += t.x; acc.y += t.y; acc.z += t.z; acc.w += t.w;
        }
        red[tid] = acc;
    }
    __syncthreads();
    if (tid < DC) {
        f4 s = red[tid];
#pragma unroll
        for (int q = 1; q < 5; ++q) {
            const f4 t = red[tid + DC * q];
            s.x += t.x; s.y += t.y; s.z += t.z; s.w += t.w;
        }
        const float sc = 1.0f / (float)LNEG;
        s.x *= sc; s.y *= sc; s.z *= sc; s.w *= sc;
        ((f4*)S)[b * DC + tid] = s;
    }
}

// ---------------------------------------------------------------------------
// Kernel 2: v[b,d'] = sum_d W_att[d',d] * ys[b,d]   (exact f32 WMMA)
// GEMM: A = ys [512 x 100], B[k=d, n=d'] = W_att[n*D + k], C = v [512 x 100].
// One wave per 16x16 output tile; K loop = 25 steps of V_WMMA_F32_16X16X4_F32.
//
// Fragment layouts per CDNA5 ISA (CDNA5 ISA):
//   A 16x4 f32: lanes 0-15 -> M, VGPR0 = {K=0 | K=2}, VGPR1 = {K=1 | K=3}
//   B 4x16 f32: lanes 0-15 -> N, mirrored K striping
//   C/D 16x16:  VGPR r: lanes 0-15 -> M=r, lanes 16-31 -> M=8+r
// ---------------------------------------------------------------------------
__global__ __launch_bounds__(32) void k_v_wmma(const float* __restrict__ ys,
                                               const float* __restrict__ W_att,
                                               float* __restrict__ v) {
    const int m_base = blockIdx.x * 16;       // 32 tiles cover M=512
    const int n_base = blockIdx.y * 16;       // 7 tiles cover N=100 (padded 112)
    const int t      = threadIdx.x;           // 0..31
    const int lane16 = t & 15;
    const int khalf  = (t >> 4) * 2;          // 0 or 2
    const int m      = m_base + lane16;
    const int n      = n_base + lane16;
    const bool nok   = (n < Dn);

    v8f c = {};
    for (int k0 = 0; k0 < Dn; k0 += 4) {
        v2f a, bm;
        a.x = ys[m * Dn + k0 + khalf];
        a.y = ys[m * Dn + k0 + khalf + 1];
        if (nok) {
            bm.x = W_att[n * Dn + k0 + khalf];      // B[k,n] = W_att[n, k]
            bm.y = W_att[n * Dn + k0 + khalf + 1];
        } else {
            bm.x = 0.0f; bm.y = 0.0f;
        }
        c = __builtin_amdgcn_wmma_f32_16x16x4_f32(
                /*neg_a=*/false, a, /*neg_b=*/false, bm,
                /*c_mod=*/(short)0, c, /*reuse_a=*/false, /*reuse_b=*/false);
    }

    if (nok) {
        const int mrow = m_base + (t >> 4) * 8;   // lanes 16-31 hold M=8..15
#pragma unroll
        for (int r = 0; r < 8; ++r) {
            v[(mrow + r) * Dn + n] = c[r];
        }
    }
}

// ---------------------------------------------------------------------------
// Kernel 3: masked attention + zs.  One block per b, 256 threads (1 per l).
// e tile staged in LDS with stride 104 floats (16B-aligned rows so both the
// global fill and the per-row dot use B128 ops).  ~109 KB dynamic LDS.
// ---------------------------------------------------------------------------
#define ESTRIDE 104
__global__ __launch_bounds__(256) void k_attn(const int* __restrict__ pos_ids,
                                              const float* __restrict__ emb,
                                              const float* __restrict__ v,
                                              float* __restrict__ zs) {
    extern __shared__ float sm[];
    float* e_sh = sm;                          // 256 * 104
    float* v_sh = e_sh + Ln * ESTRIDE;         // 104 (padded; 16B aligned)
    float* f_sh = v_sh + 104;                  // 256 (att weights)
    float* red  = f_sh + Ln;                   // 256 (reduction scratch)

    const int b   = blockIdx.x;
    const int tid = threadIdx.x;

    if (tid < Dn) v_sh[tid] = v[b * Dn + tid];

    // cooperative, coalesced B128 fill of the 256x100 e tile
    const int* pid  = pos_ids + b * Ln;
    const f4*  emb4 = (const f4*)emb;
    for (int idx = tid; idx < Ln * DC; idx += 256) {
        const int l = idx / DC;
        const int c = idx - l * DC;
        *(f4*)(e_sh + l * ESTRIDE + 4 * c) = emb4[pid[l] * DC + c];
    }
    __syncthreads();

    // f[l] = mask ? e[l,:]·v : -1e9   (B128 LDS reads)
    const int myid = pid[tid];
    float f;
    if (myid != 0) {
        const f4* erow = (const f4*)(e_sh + tid * ESTRIDE);
        const f4* v4   = (const f4*)v_sh;
        float acc = 0.0f;
#pragma unroll 5
        for (int c = 0; c < DC; ++c) {
            const f4 e = erow[c];
            const f4 w = v4[c];
            acc += e.x * w.x + e.y * w.y + e.z * w.z + e.w * w.w;
        }
        f = acc;
    } else {
        f = -1e9f;
    }

    // block max
    red[tid] = f;
    __syncthreads();
    for (int s = 128; s > 0; s >>= 1) {
        if (tid < s) red[tid] = fmaxf(red[tid], red[tid + s]);
        __syncthreads();
    }
    const float mx = red[0];
    __syncthreads();

    // block sum of exp
    const float ex = __expf(f - mx);
    red[tid] = ex;
    __syncthreads();
    for (int s = 128; s > 0; s >>= 1) {
        if (tid < s) red[tid] += red[tid + s];
        __syncthreads();
    }
    const float inv = 1.0f / red[0];
    __syncthreads();

    f_sh[tid] = ex * inv;                      // att[l]
    __syncthreads();

    // zs[b,d] = sum_l att[l] * e[l,d]
    if (tid < Dn) {
        float acc = 0.0f;
        for (int l = 0; l < Ln; ++l) acc += f_sh[l] * e_sh[l * ESTRIDE + tid];
        zs[b * Dn + tid] = acc;
    }
}

// ---------------------------------------------------------------------------
// Kernel 4: pt = sigmoid(zs@Wk + bk), rs = pt@Wf + bf, per-b partial sums of
// pos_term (rs·zs) and neg_term (rs·S).  One block per b.
// ---------------------------------------------------------------------------
__global__ __launch_bounds__(128) void k_head(const float* __restrict__ zs,
                                              const float* __restrict__ S,
                                              const float* __restrict__ Wk,
                                              const float* __restrict__ bk,
                                              const float* __restrict__ Wf,
                                              const float* __restrict__ bf,
                                              float* __restrict__ out_pt,
                                              float* __restrict__ out_rs,
                                              float* __restrict__ pos_part,
                                              float* __restrict__ neg_part) {
    __shared__ float zs_sh[Dn];
    __shared__ float pt_sh[16];
    __shared__ float red[128];
    const int b   = blockIdx.x;
    const int tid = threadIdx.x;

    if (tid < Dn) zs_sh[tid] = zs[b * Dn + tid];
    __syncthreads();

    if (tid < Kn) {
        float acc = bk[tid];
        for (int d = 0; d < Dn; ++d) acc += zs_sh[d] * Wk[d * Kn + tid];
        const float p = 1.0f / (1.0f + __expf(-acc));
        pt_sh[tid] = p;
        out_pt[b * Kn + tid] = p;
    }
    __syncthreads();

    float lpos = 0.0f, lneg = 0.0f;
    if (tid < Dn) {
        float acc = bf[tid];
#pragma unroll
        for (int k = 0; k < Kn; ++k) acc += pt_sh[k] * Wf[k * Dn + tid];
        out_rs[b * Dn + tid] = acc;
        lpos = acc * zs_sh[tid];
        lneg = acc * S[b * Dn + tid];
    }

    red[tid] = lpos;
    __syncthreads();
    for (int s = 64; s > 0; s >>= 1) { if (tid < s) red[tid] += red[tid + s]; __syncthreads(); }
    if (tid == 0) pos_part[b] = red[0];
    __syncthreads();

    red[tid] = lneg;
    __syncthreads();
    for (int s = 64; s > 0; s >>= 1) { if (tid < s) red[tid] += red[tid + s]; __syncthreads(); }
    if (tid == 0) neg_part[b] = red[0];
}

// ---------------------------------------------------------------------------
// Kernel 5: reg + final loss.  Single block, fixed-order reductions.
// ---------------------------------------------------------------------------
__global__ __launch_bounds__(256) void k_final(const float* __restrict__ Wf,
                                               const float* __restrict__ pos_part,
                                               const float* __restrict__ neg_part,
                                               float* __restrict__ out_loss) {
    __shared__ float norms[Kn];
    __shared__ float red[256];
    const int tid = threadIdx.x;

    if (tid < Kn) {
        float acc = 0.0f;
        for (int d = 0; d < Dn; ++d) { const float w = Wf[tid * Dn + d]; acc += w * w; }
        norms[tid] = sqrtf(acc);
    }
    __syncthreads();

    float r = 0.0f;
    if (tid < Kn * Kn) {
        const int i = tid / Kn;
        const int j = tid - i * Kn;
        const float x = norms[i] * norms[j] - 1.0f;
        r = x * x;
    }
    red[tid] = r;
    __syncthreads();
    for (int s = 128; s > 0; s >>= 1) { if (tid < s) red[tid] += red[tid + s]; __syncthreads(); }
    const float reg = sqrtf(red[0]);
    __syncthreads();

    float p = 0.0f;
    for (int i = tid; i < Bn; i += 256) p += pos_part[i];
    red[tid] = p;
    __syncthreads();
    for (int s = 128; s > 0; s >>= 1) { if (tid < s) red[tid] += red[tid + s]; __syncthreads(); }
    const float pos = red[0];
    __syncthreads();

    float q = 0.0f;
    for (int i = tid; i < Bn; i += 256) q += neg_part[i];
    red[tid] = q;
    __syncthreads();
    for (int s = 128; s > 0; s >>= 1) { if (tid < s) red[tid] += red[tid + s]; __syncthreads(); }

    if (tid == 0) {
        const float neg = red[0];
        const float margin = 1.0f - neg + pos;
        out_loss[0] = fmaxf(margin, 0.0f) + 1.0f /*LAMBDA*/ * reg;
    }
}

// ---------------------------------------------------------------------------
extern "C" void kernel_launch(void* const* d_in, const int* in_sizes, int n_in,
                              void* d_out, int out_size, void* d_ws, size_t ws_size,
                              hipStream_t stream) {
    const int*   pos_ids = (const int*)d_in[0];    // [512,256]
    const int*   neg_ids = (const int*)d_in[1];    // [512,20,64]
    const float* emb     = (const float*)d_in[2];  // [50001,100]
    const float* W_att   = (const float*)d_in[3];  // [100,100]
    const float* Wk      = (const float*)d_in[4];  // [100,14]
    const float* bk      = (const float*)d_in[5];  // [14]
    const float* Wf      = (const float*)d_in[6];  // [14,100]
    const float* bf      = (const float*)d_in[7];  // [100]

    float* out      = (float*)d_out;
    float* out_loss = out;                          // [1]
    float* out_pt   = out + 1;                      // [512,14]
    float* out_rs   = out + 1 + Bn * Kn;            // [512,100]

    float* ws       = (float*)d_ws;
    float* ys       = ws;                           // 51200
    float* S        = ys + Bn * Dn;                 // 51200
    float* v        = S + Bn * Dn;                  // 51200
    float* zs       = v + Bn * Dn;                  // 51200
    float* pos_part = zs + Bn * Dn;                 // 512
    float* neg_part = pos_part + Bn;                // 512

    k_gather<<<Bn, 128, 0, stream>>>(pos_ids, neg_ids, emb, ys, S);

    k_v_wmma<<<dim3(Bn / 16, (Dn + 15) / 16), 32, 0, stream>>>(ys, W_att, v);

    const size_t attn_lds = (size_t)(Ln * ESTRIDE + 104 + Ln + Ln) * sizeof(float);
    k_attn<<<Bn, 256, attn_lds, stream>>>(pos_ids, emb, v, zs);

    k_head<<<Bn, 128, 0, stream>>>(zs, S, Wk, bk, Wf, bf,
                                   out_pt, out_rs, pos_part, neg_part);

    k_final<<<1, 256, 0, stream>>>(Wf, pos_part, neg_part, out_loss);
}